// MeshPotential_8349416423818
// MI455X (gfx1250) — compile-verified
//
#include <hip/hip_runtime.h>
#include <math.h>

// ---------------------------------------------------------------------------
// P3M / PME potential on a 128^3 mesh, 2 species channels, for gfx1250.
//
// Pipeline:
//   1) setup: invert 3x3 cell, store inv + |det| in workspace
//   2) twiddle tables: W_fwd and W_inv 128x128 (cos/sin), f32
//   3) zero complex mesh (ch0 -> real plane, ch1 -> imag plane)
//   4) scatter: order-4 separable stencil, f32 global atomics
//   5) 3 forward DFT passes (z,y,x): complex GEMM via V_WMMA_F32_16X16X4_F32,
//      compile-time strides, 4 line-tiles per wave (A-reuse)
//   6) pointwise G(k) multiply (real, even kernel; includes 1/volume)
//   7) 3 inverse DFT passes
//   8) gather: same stencil; out[atom][0]=Re, out[atom][1]=Im
// ---------------------------------------------------------------------------

#define NSM   128
#define NSM2  (NSM * NSM)          // 16384
#define MESH_N (NSM * NSM * NSM)   // 2097152

__device__ __constant__ float kPI = 3.14159265358979323846f;

typedef float v2f __attribute__((ext_vector_type(2)));
typedef float v8f __attribute__((ext_vector_type(8)));

// ------------------------------ setup --------------------------------------
__global__ void pme_setup(const float* __restrict__ cell, float* __restrict__ misc) {
    if (threadIdx.x != 0 || blockIdx.x != 0) return;
    float a = cell[0], b = cell[1], c = cell[2];
    float d = cell[3], e = cell[4], f = cell[5];
    float g = cell[6], h = cell[7], i = cell[8];
    float det = a * (e * i - f * h) - b * (d * i - f * g) + c * (d * h - e * g);
    float invdet = 1.0f / det;
    misc[0] =  (e * i - f * h) * invdet;
    misc[1] = -(b * i - c * h) * invdet;
    misc[2] =  (b * f - c * e) * invdet;
    misc[3] = -(d * i - f * g) * invdet;
    misc[4] =  (a * i - c * g) * invdet;
    misc[5] = -(a * f - c * d) * invdet;
    misc[6] =  (d * h - e * g) * invdet;
    misc[7] = -(a * h - b * g) * invdet;
    misc[8] =  (a * e - b * d) * invdet;
    misc[9] = fabsf(det);   // volume
}

// --------------------------- twiddle tables --------------------------------
__global__ void pme_twiddle(float* __restrict__ wfRe, float* __restrict__ wfIm,
                            float* __restrict__ wiRe, float* __restrict__ wiIm) {
    int i = blockIdx.x * blockDim.x + threadIdx.x;   // 0 .. 16383
    int k = i >> 7, n = i & 127;
    int p = (k * n) & 127;                           // exact phase reduction
    float ang = -2.0f * kPI * (float)p * (1.0f / 128.0f);
    float s, c;
    sincosf(ang, &s, &c);
    wfRe[i] = c;  wfIm[i] = s;      // forward:  e^{-i 2pi kn/N}
    wiRe[i] = c;  wiIm[i] = -s;     // inverse:  e^{+i 2pi kn/N}, unscaled
}

// ------------------------------ zero mesh ----------------------------------
__global__ void pme_zero(float* __restrict__ p, int n) {
    int i = blockIdx.x * blockDim.x + threadIdx.x;
    if (i < n) p[i] = 0.0f;
}

// ------------------------- P3M order-4 weights -----------------------------
__device__ __forceinline__ void p3m_weights(float x, float w[4]) {
    float x2 = x * x, x3 = x2 * x;
    const float s = 1.0f / 48.0f;
    w[0] = (1.0f  - 6.0f  * x + 12.0f * x2 - 8.0f  * x3) * s;
    w[1] = (23.0f - 30.0f * x - 12.0f * x2 + 24.0f * x3) * s;
    w[2] = (23.0f + 30.0f * x - 12.0f * x2 - 24.0f * x3) * s;
    w[3] = (1.0f  + 6.0f  * x + 12.0f * x2 + 8.0f  * x3) * s;
}

__device__ __forceinline__ void p3m_stencil(const float* __restrict__ pos,
                                            const float* __restrict__ misc,
                                            int i, int idx[3][4], float w[3][4]) {
    float px = pos[3 * i], py = pos[3 * i + 1], pz = pos[3 * i + 2];
    #pragma unroll
    for (int d = 0; d < 3; ++d) {
        // pos_rel = ns * (positions @ inv_cell)
        float pr = (float)NSM * (px * misc[d] + py * misc[3 + d] + pz * misc[6 + d]);
        float bf = floorf(pr);
        float off = pr - bf - 0.5f;
        float wd[4];
        p3m_weights(off, wd);
        int b = (int)bf;
        #pragma unroll
        for (int t = 0; t < 4; ++t) {
            idx[d][t] = (b + t - 1) & (NSM - 1);   // mod 128 (handles -1 wrap)
            w[d][t] = wd[t];
        }
    }
}

// ------------------------------ scatter ------------------------------------
__global__ void pme_scatter(const float* __restrict__ pos, const int* __restrict__ species,
                            const float* __restrict__ misc,
                            float* __restrict__ meshRe, float* __restrict__ meshIm,
                            int nAtoms) {
    int i = blockIdx.x * blockDim.x + threadIdx.x;
    if (i >= nAtoms) return;
    int idx[3][4];
    float w[3][4];
    p3m_stencil(pos, misc, i, idx, w);
    // one-hot charge: channel 0 -> real plane, channel 1 -> imag plane
    float* mesh = (species[i] == 0) ? meshRe : meshIm;
    #pragma unroll
    for (int l = 0; l < 4; ++l) {
        #pragma unroll
        for (int m = 0; m < 4; ++m) {
            float wlm = w[0][l] * w[1][m];
            int o = idx[0][l] * NSM2 + idx[1][m] * NSM;
            #pragma unroll
            for (int n = 0; n < 4; ++n)
                atomicAdd(&mesh[o + idx[2][n]], wlm * w[2][n]);
        }
    }
}

// ------------------- DFT pass = complex GEMM via f32 WMMA ------------------
// Y[k, line] = sum_n W[k,n] * X[n, line].
// Element address = a*SA + b*SB + n*SN with line L = a*128 + b (all strides
// compile-time so loads vectorize / fold into immediate offsets).
// Each wave computes one 16-row k-tile x NT=4 adjacent 16-column line-tiles,
// reusing the A (twiddle) registers across the 4 B tiles.
template <int SA, int SB, int SN>
__global__ void __launch_bounds__(256)
pme_dft_pass(const float* __restrict__ srcRe, const float* __restrict__ srcIm,
             float* __restrict__ dstRe, float* __restrict__ dstIm,
             const float* __restrict__ wRe, const float* __restrict__ wIm) {
    constexpr int NT = 4;                          // line-tiles per wave
    int lane = threadIdx.x & 31;
    int g = blockIdx.x * 8 + (threadIdx.x >> 5);   // global wave id, 0..2047
    int kTile     = (g & 7) << 4;                  // 8 k-tiles
    int lineGroup = (g >> 3) << 6;                 // 256 groups of 64 lines
    int row = lane & 15;
    int hi  = lane >> 4;

    int colBase[NT];
    #pragma unroll
    for (int t = 0; t < NT; ++t) {
        int L = lineGroup + t * 16 + row;          // this lane's column, tile t
        colBase[t] = (L >> 7) * SA + (L & 127) * SB;
    }
    int wrow = (kTile + row) << 7;                 // W row base (row-major 128)

    v8f cr[NT], ci[NT];
    #pragma unroll
    for (int t = 0; t < NT; ++t) { cr[t] = (v8f){}; ci[t] = (v8f){}; }

    for (int kc = 0; kc < 32; ++kc) {
        int n0 = (kc << 2) + (hi << 1);            // K-chunk, ISA 16x4 A layout
        v2f ar, ai, nai;
        ar.x = wRe[wrow + n0];     ar.y = wRe[wrow + n0 + 1];
        ai.x = wIm[wrow + n0];     ai.y = wIm[wrow + n0 + 1];
        nai.x = -ai.x;             nai.y = -ai.y;  // f32 WMMA has no A/B NEG

        #pragma unroll
        for (int t = 0; t < NT; ++t) {
            int o0 = colBase[t] + n0 * SN;
            v2f br, bi;
            br.x = srcRe[o0];      br.y = srcRe[o0 + SN];
            bi.x = srcIm[o0];      bi.y = srcIm[o0 + SN];
            // Cr += Ar*Br - Ai*Bi ;  Ci += Ar*Bi + Ai*Br
            cr[t] = __builtin_amdgcn_wmma_f32_16x16x4_f32(false, ar,  false, br, (short)0, cr[t], false, false);
            cr[t] = __builtin_amdgcn_wmma_f32_16x16x4_f32(false, nai, false, bi, (short)0, cr[t], false, false);
            ci[t] = __builtin_amdgcn_wmma_f32_16x16x4_f32(false, ar,  false, bi, (short)0, ci[t], false, false);
            ci[t] = __builtin_amdgcn_wmma_f32_16x16x4_f32(false, ai,  false, br, (short)0, ci[t], false, false);
        }
    }

    // D layout: VGPR v, lanes 0-15 -> M=v, lanes 16-31 -> M=v+8 (same N=row)
    #pragma unroll
    for (int t = 0; t < NT; ++t) {
        #pragma unroll
        for (int v = 0; v < 8; ++v) {
            int k = kTile + v + (hi << 3);
            int o = colBase[t] + k * SN;
            dstRe[o] = cr[t][v];
            dstIm[o] = ci[t][v];
        }
    }
}

// ------------------------- G(k) pointwise multiply -------------------------
__global__ void pme_gmult(float* __restrict__ re, float* __restrict__ im,
                          const float* __restrict__ misc) {
    int i = blockIdx.x * blockDim.x + threadIdx.x;   // 0 .. MESH_N-1
    int x = i >> 14, y = (i >> 7) & 127, z = i & 127;
    float fx = (float)(x < 64 ? x : x - 128);
    float fy = (float)(y < 64 ? y : y - 128);
    float fz = (float)(z < 64 ? z : z - 128);
    const float TWO_PI = 6.28318530717958647692f;
    // k_j = 2*pi * sum_d f_d * inv_cell[j][d]
    float kx = TWO_PI * (fx * misc[0] + fy * misc[1] + fz * misc[2]);
    float ky = TWO_PI * (fx * misc[3] + fy * misc[4] + fz * misc[5]);
    float kz = TWO_PI * (fx * misc[6] + fy * misc[7] + fz * misc[8]);
    float ksq = kx * kx + ky * ky + kz * kz;
    float G = 0.0f;
    const float smearing = 1.0f;
    if (ksq > 0.0f)
        G = 4.0f * kPI * expf(-0.5f * smearing * smearing * ksq) / (ksq * misc[9]);
    re[i] *= G;
    im[i] *= G;
}

// ------------------------------ gather -------------------------------------
__global__ void pme_gather(const float* __restrict__ pos, const float* __restrict__ misc,
                           const float* __restrict__ meshRe, const float* __restrict__ meshIm,
                           float* __restrict__ out, int nAtoms) {
    int i = blockIdx.x * blockDim.x + threadIdx.x;
    if (i >= nAtoms) return;
    int idx[3][4];
    float w[3][4];
    p3m_stencil(pos, misc, i, idx, w);
    float p0 = 0.0f, p1 = 0.0f;
    #pragma unroll
    for (int l = 0; l < 4; ++l) {
        #pragma unroll
        for (int m = 0; m < 4; ++m) {
            float wlm = w[0][l] * w[1][m];
            int o = idx[0][l] * NSM2 + idx[1][m] * NSM;
            #pragma unroll
            for (int n = 0; n < 4; ++n) {
                float w3 = wlm * w[2][n];
                int oo = o + idx[2][n];
                p0 += w3 * meshRe[oo];
                p1 += w3 * meshIm[oo];
            }
        }
    }
    out[2 * i + 0] = p0;
    out[2 * i + 1] = p1;
}

// ------------------------------ launch -------------------------------------
extern "C" void kernel_launch(void* const* d_in, const int* in_sizes, int n_in,
                              void* d_out, int out_size, void* d_ws, size_t ws_size,
                              hipStream_t stream) {
    const float* positions = (const float*)d_in[0];   // (N,3) f32
    const float* cell      = (const float*)d_in[1];   // (3,3) f32
    const int*   species   = (const int*)d_in[2];     // (N,)  i32
    // d_in[3] = ns (compile-time 128)
    const int nAtoms = in_sizes[2];

    // Workspace layout (f32 units):
    //   [0]        meshA_re  (MESH_N)
    //   [1*M]      meshA_im
    //   [2*M]      meshB_re
    //   [3*M]      meshB_im
    //   [4*M]      Wf_re, Wf_im, Wi_re, Wi_im (4 x 16384)
    //   [...]      misc: inv_cell[9], volume
    float* ws = (float*)d_ws;
    const size_t M = (size_t)MESH_N;
    float* aRe  = ws;
    float* aIm  = ws + M;
    float* bRe  = ws + 2 * M;
    float* bIm  = ws + 3 * M;
    float* wfRe = ws + 4 * M;
    float* wfIm = wfRe + NSM2;
    float* wiRe = wfIm + NSM2;
    float* wiIm = wiRe + NSM2;
    float* misc = wiIm + NSM2;

    float* out = (float*)d_out;

    pme_setup<<<1, 32, 0, stream>>>(cell, misc);
    pme_twiddle<<<NSM2 / 256, 256, 0, stream>>>(wfRe, wfIm, wiRe, wiIm);
    pme_zero<<<(2 * MESH_N) / 256, 256, 0, stream>>>(aRe, 2 * MESH_N);

    pme_scatter<<<(nAtoms + 255) / 256, 256, 0, stream>>>(
        positions, species, misc, aRe, aIm, nAtoms);

    // 2048 waves per pass = 8 k-tiles x 256 line-groups (4 tiles each)
    const int DFT_BLOCKS = 256;

    // Forward FFT (unscaled): z, y, x
    pme_dft_pass<NSM2, NSM, 1><<<DFT_BLOCKS, 256, 0, stream>>>(
        aRe, aIm, bRe, bIm, wfRe, wfIm);                       // along z
    pme_dft_pass<NSM2, 1, NSM><<<DFT_BLOCKS, 256, 0, stream>>>(
        bRe, bIm, aRe, aIm, wfRe, wfIm);                       // along y
    pme_dft_pass<NSM, 1, NSM2><<<DFT_BLOCKS, 256, 0, stream>>>(
        aRe, aIm, bRe, bIm, wfRe, wfIm);                       // along x

    pme_gmult<<<MESH_N / 256, 256, 0, stream>>>(bRe, bIm, misc);

    // Inverse FFT (unscaled, norm="forward" convention): z, y, x
    pme_dft_pass<NSM2, NSM, 1><<<DFT_BLOCKS, 256, 0, stream>>>(
        bRe, bIm, aRe, aIm, wiRe, wiIm);
    pme_dft_pass<NSM2, 1, NSM><<<DFT_BLOCKS, 256, 0, stream>>>(
        aRe, aIm, bRe, bIm, wiRe, wiIm);
    pme_dft_pass<NSM, 1, NSM2><<<DFT_BLOCKS, 256, 0, stream>>>(
        bRe, bIm, aRe, aIm, wiRe, wiIm);

    pme_gather<<<(nAtoms + 255) / 256, 256, 0, stream>>>(
        positions, misc, aRe, aIm, out, nAtoms);
}